// BaseDiscretGenerator_5703716569694
// MI455X (gfx1250) — compile-verified
//
#include <hip/hip_runtime.h>

// BaseDiscretGenerator: bucketize fake_tensor against per-column cumsum
// boundaries. Memory-bound (≈243 MB traffic → ~10.4 µs @ 23.3 TB/s), so the
// kernel is a streaming pipeline: per-column boundaries live in registers,
// the fake_tensor stream is double-buffered through LDS with the gfx1250
// async DMA path (global_load_async_to_lds_b32 / s_wait_asynccnt), and the
// one-hot dist rows are emitted as coalesced float2 stores. WMMA is
// deliberately NOT used: there are zero matmul FLOPs in this op.

#define EPSV 1e-4f
#define TILE_D 256
#define ROWS_PER_BLOCK 64

__global__ __launch_bounds__(TILE_D)
void discret_gen_kernel(const float* __restrict__ fake,   // [B,D]
                        const float* __restrict__ minb,   // [D]
                        const float* __restrict__ lens,   // [D,4]
                        float* __restrict__ out,          // dist [B,D,6] ++ val [B,D]
                        int B, int D)
{
    __shared__ float smem[2][TILE_D];

    const int t  = threadIdx.x;
    const int d  = blockIdx.x * TILE_D + t;
    const int dc = (d < D) ? d : (D - 1);     // clamped: loads stay in-bounds
    const bool ok = (d < D);

    const int r0 = blockIdx.y * ROWS_PER_BLOCK;
    int r1 = r0 + ROWS_PER_BLOCK; if (r1 > B) r1 = B;
    const int n = r1 - r0;
    if (n <= 0) return;

    // ---- per-column boundaries, computed once and held in registers ----
    const float4 L = *(const float4*)(lens + (size_t)dc * 4);  // 16B aligned
    const float b0 = minb[dc];
    const float b1 = b0 + fmaxf(L.x, 0.f) + EPSV;
    const float b2 = b1 + fmaxf(L.y, 0.f) + EPSV;
    const float b3 = b2 + fmaxf(L.z, 0.f) + EPSV;
    const float b4 = b3 + fmaxf(L.w, 0.f) + EPSV;

    // Workgroup-relative LDS byte offsets for the two staging buffers.
    // Low 32 bits of the flat shared-aperture address == LDS offset
    // (aperture base occupies bits 63:32), which is what the async op's
    // per-lane VDST address expects (HW adds LDS_BASE).
    const unsigned lds0 = (unsigned)(unsigned long long)(void*)&smem[0][t];
    const unsigned lds1 = (unsigned)(unsigned long long)(void*)&smem[1][t];

    // Prime the pipeline: async DMA row r0 -> buffer 0 (tracked by ASYNCcnt).
    {
        const unsigned goff = (unsigned)(((size_t)r0 * D + dc) * sizeof(float));
        asm volatile("global_load_async_to_lds_b32 %0, %1, %2"
                     :: "v"(lds0), "v"(goff), "s"(fake) : "memory");
    }

    float* __restrict__ dist = out;
    float* __restrict__ val  = out + (size_t)B * D * 6;

    for (int i = 0; i < n; ++i) {
        const int row = r0 + i;

        if (i + 1 < n) {
            // Issue next row's async copy into the other buffer, then wait
            // until only that one copy is outstanding -> current row landed.
            const unsigned nxt  = ((i + 1) & 1) ? lds1 : lds0;
            const unsigned goff = (unsigned)(((size_t)(row + 1) * D + dc) * sizeof(float));
            asm volatile("global_load_async_to_lds_b32 %0, %1, %2"
                         :: "v"(nxt), "v"(goff), "s"(fake) : "memory");
            if (i + 4 < n)   // pre-touch the stream deeper into L2
                __builtin_prefetch(fake + (size_t)(row + 4) * D + dc, 0, 0);
            asm volatile("s_wait_asynccnt 0x1" ::: "memory");
        } else {
            asm volatile("s_wait_asynccnt 0x0" ::: "memory");
        }

        const float a = smem[i & 1][t];

        // cnt = number of boundaries strictly below a; equality with any
        // boundary makes H(0)=0 kill every product term -> all-zero row.
        const int  cnt = (a > b0) + (a > b1) + (a > b2) + (a > b3) + (a > b4);
        const bool ne  = (a != b0) & (a != b1) & (a != b2) & (a != b3) & (a != b4);

        if (ok) {
            const size_t e = (size_t)row * D + d;
            float2* dp = (float2*)(dist + e * 6);   // byte offset 24*e: 8B aligned
            float2 v01, v23, v45;
            v01.x = (ne && cnt == 0) ? 1.f : 0.f;
            v01.y = (ne && cnt == 1) ? 1.f : 0.f;
            v23.x = (ne && cnt == 2) ? 1.f : 0.f;
            v23.y = (ne && cnt == 3) ? 1.f : 0.f;
            v45.x = (ne && cnt == 4) ? 1.f : 0.f;
            v45.y = (ne && cnt == 5) ? 1.f : 0.f;
            dp[0] = v01; dp[1] = v23; dp[2] = v45;
            val[e] = ne ? (float)cnt : 0.f;
        }
    }
}

extern "C" void kernel_launch(void* const* d_in, const int* in_sizes, int n_in,
                              void* d_out, int out_size, void* d_ws, size_t ws_size,
                              hipStream_t stream) {
    const float* fake = (const float*)d_in[0];   // [B,D]
    const float* minb = (const float*)d_in[1];   // [D]
    const float* lens = (const float*)d_in[2];   // [D,4]
    const int D = in_sizes[1];
    const int B = in_sizes[0] / D;

    dim3 block(TILE_D);
    dim3 grid((D + TILE_D - 1) / TILE_D, (B + ROWS_PER_BLOCK - 1) / ROWS_PER_BLOCK);
    hipLaunchKernelGGL(discret_gen_kernel, grid, block, 0, stream,
                       fake, minb, lens, (float*)d_out, B, D);
}